// CustomLoss_4363686773577
// MI455X (gfx1250) — compile-verified
//
#include <hip/hip_runtime.h>
#include <math.h>

#define GRID_N 49
#define BB 2
#define NCLS 20
#define FEAT 30       // 5*BB + NCLS
#define NOBJ 10
#define LAMBDA_C 5.0f
#define LAMBDA_N 0.5f
#define RED_CHUNK 128 // elements consumed per reduce-loop iteration (2 WMMAs)

typedef __attribute__((ext_vector_type(2))) float v2f;
typedef __attribute__((ext_vector_type(8))) float v8f;

// ---------------------------------------------------------------------------
// Kernel 1: one block per batch item. Stage outputs[b] (1470 f32) and the
// transformed target[b] through LDS (coalesced), then thread c (< 49) computes
// the full per-cell loss. Block-reduce to a per-batch partial in d_ws.
// ---------------------------------------------------------------------------
__global__ void __launch_bounds__(64)
yolo_loss_main(const float* __restrict__ outputs,
               const float* __restrict__ target,
               float* __restrict__ partials,
               int batch)
{
    __shared__ float s_out[GRID_N * FEAT];  // 1470
    __shared__ float s_tgt[NOBJ * 5];       // 50 (post sqrt / nan_to_num)
    __shared__ int   s_cell[NOBJ];
    __shared__ int   s_valid[NOBJ];
    __shared__ float s_red[64];

    const int b   = blockIdx.x;
    const int tid = threadIdx.x;

    // coalesced stage of outputs[b]
    const float* ob = outputs + (size_t)b * (GRID_N * FEAT);
    for (int i = tid; i < GRID_N * FEAT; i += 64)
        s_out[i] = ob[i];

    // target[b]: sqrt on w,h then nan -> -1.0 (nan_to_num)
    const float* tb = target + (size_t)b * (NOBJ * 5);
    for (int i = tid; i < NOBJ * 5; i += 64) {
        float v = tb[i];
        int d = i % 5;
        if (d == 2 || d == 3) v = sqrtf(v);
        if (isnan(v)) v = -1.0f;
        s_tgt[i] = v;
    }
    __syncthreads();

    if (tid < NOBJ) {
        float xs = s_tgt[tid * 5 + 0];
        float ys = s_tgt[tid * 5 + 1];
        int xi = (int)truncf(7.0f * xs);
        int yi = (int)truncf(7.0f * ys);
        int cell = 7 * yi + xi;
        cell = min(max(cell, 0), GRID_N - 1);
        s_cell[tid]  = cell;
        s_valid[tid] = ((xs != -1.0f) && (xi >= 0)) ? 1 : 0;
    }
    __syncthreads();

    float loss = 0.0f;

    if (tid < GRID_N) {
        const int c = tid;

        // ---- build gt[2][5]: ordered slot assignment (matches cumsum/one_hot) ----
        float gt[BB][5];
        #pragma unroll
        for (int s = 0; s < BB; ++s)
            #pragma unroll
            for (int d = 0; d < 5; ++d) gt[s][d] = 0.0f;

        int cnt = 0;
        for (int o = 0; o < NOBJ; ++o) {
            if (s_valid[o] && s_cell[o] == c) {
                float ssum = s_tgt[o*5+0] + s_tgt[o*5+1] + s_tgt[o*5+2]
                           + s_tgt[o*5+3] + s_tgt[o*5+4];
                if (ssum > 0.0f) {
                    if (cnt < BB) {
                        #pragma unroll
                        for (int d = 0; d < 5; ++d) gt[cnt][d] = s_tgt[o*5+d];
                    }
                    cnt++;  // extras clamp to slot BB and are dropped
                }
            }
        }

        // ---- pred boxes ----
        float pred[BB][5];
        #pragma unroll
        for (int p = 0; p < BB; ++p)
            #pragma unroll
            for (int d = 0; d < 5; ++d)
                pred[p][d] = s_out[c * FEAT + 5 * p + d];

        // ---- IoU(gt, pred) : corners use (sqrt-domain value)^2 ----
        float gmn[BB][2], gmx[BB][2], garea[BB];
        #pragma unroll
        for (int g = 0; g < BB; ++g) {
            #pragma unroll
            for (int j = 0; j < 2; ++j) {
                float wh = gt[g][2 + j];          // already sqrt(target wh)
                wh = wh * wh;
                gmn[g][j] = gt[g][j] - wh;
                gmx[g][j] = gt[g][j] + wh;
            }
            garea[g] = fabsf(gmx[g][0] - gmn[g][0]) * fabsf(gmx[g][1] - gmn[g][1]);
        }
        float pmn[BB][2], pmx[BB][2], parea[BB];
        #pragma unroll
        for (int p = 0; p < BB; ++p) {
            #pragma unroll
            for (int j = 0; j < 2; ++j) {
                float sw = sqrtf(pred[p][2 + j]); // pdims sqrt, then squared
                float wh = sw * sw;
                pmn[p][j] = pred[p][j] - wh;
                pmx[p][j] = pred[p][j] + wh;
            }
            parea[p] = fabsf(pmx[p][0] - pmn[p][0]) * fabsf(pmx[p][1] - pmn[p][1]);
        }

        int tindex[BB];
        #pragma unroll
        for (int g = 0; g < BB; ++g) {
            float iouv[BB];
            #pragma unroll
            for (int p = 0; p < BB; ++p) {
                float d0 = fminf(gmx[g][0], pmx[p][0]) - fmaxf(gmn[g][0], pmn[p][0]);
                float d1 = fminf(gmx[g][1], pmx[p][1]) - fmaxf(gmn[g][1], pmn[p][1]);
                d0 = fmaxf(d0, 0.0f);
                d1 = fmaxf(d1, 0.0f);
                float inter = d0 * d1;
                float un = garea[g] + parea[p] - inter;
                iouv[p] = (un == 0.0f) ? 0.0f : (inter / un);
            }
            tindex[g] = (iouv[1] > iouv[0]) ? 1 : 0;  // argmax over p, tie->first
        }

        // ---- gathered gt + truncated-IoU confidence ----
        float fg[BB][5];
        #pragma unroll
        for (int s = 0; s < BB; ++s)
            #pragma unroll
            for (int d = 0; d < 5; ++d)
                fg[s][d] = gt[tindex[s]][d];

        float confv[BB];
        #pragma unroll
        for (int s = 0; s < BB; ++s) {
            float r1[2], r2[2];
            #pragma unroll
            for (int j = 0; j < 2; ++j) {
                r1[j] = pred[s][2 + j] * pred[s][2 + j];  // raw pred wh squared
                r2[j] = fg[s][2 + j] * fg[s][2 + j];      // sqrt(t wh) squared
            }
            float d0 = fminf(pred[s][0] + 0.5f * r1[0], fg[s][0] + 0.5f * r2[0])
                     - fmaxf(pred[s][0] - 0.5f * r1[0], fg[s][0] - 0.5f * r2[0]);
            float d1 = fminf(pred[s][1] + 0.5f * r1[1], fg[s][1] + 0.5f * r2[1])
                     - fmaxf(pred[s][1] - 0.5f * r1[1], fg[s][1] - 0.5f * r2[1]);
            d0 = fmaxf(d0, 0.0f);
            d1 = fmaxf(d1, 0.0f);
            float inter2 = d0 * d1;
            float union2 = r1[0] * r1[1] + r2[0] * r2[1] - inter2;
            float ratio  = inter2 / ((union2 == 0.0f) ? 1.0f : union2);
            confv[s] = (truncf(union2) == 0.0f || truncf(inter2) == 0.0f)
                         ? 0.0f : truncf(ratio);
            fg[s][4] = confv[s];
        }

        // ---- obj loss: w = [5,5,5,5,1] ----
        #pragma unroll
        for (int s = 0; s < BB; ++s)
            #pragma unroll
            for (int d = 0; d < 5; ++d) {
                float diff = fg[s][d] - pred[s][d];
                float w = (d < 4) ? LAMBDA_C : 1.0f;
                loss += w * diff * diff;
            }

        // ---- noobj loss (mask only on last batch item, slot tindex[BB-1]) ----
        #pragma unroll
        for (int s = 0; s < BB; ++s) {
            float m = 1.0f;
            if (b == batch - 1 && s == tindex[BB - 1]) m = 0.0f;
            float e = (pred[s][4] - confv[s]) * m;
            loss += LAMBDA_N * e * e;
        }

        // ---- class loss ----
        float maxv = s_out[c * FEAT + 5 * BB];
        int amax = 0;
        for (int k = 1; k < NCLS; ++k) {
            float v = s_out[c * FEAT + 5 * BB + k];
            if (v > maxv) { maxv = v; amax = k; }  // tie -> first (argmax)
        }
        #pragma unroll
        for (int s = 0; s < BB; ++s) {
            float predsn = maxv * pred[s][4];
            float truth  = (fg[s][4] == (float)amax) ? 1.0f : 0.0f;
            float ssum   = fg[s][0] + fg[s][1] + fg[s][2] + fg[s][3] + fg[s][4];
            float cm     = (ssum > 0.0f) ? 1.0f : ((ssum < 0.0f) ? -1.0f : 0.0f);
            float diff   = truth - predsn;
            loss += cm * diff * diff;
        }
    }

    // ---- deterministic block reduction (64 threads) ----
    s_red[tid] = loss;
    __syncthreads();
    #pragma unroll
    for (int off = 32; off > 0; off >>= 1) {
        if (tid < off) s_red[tid] += s_red[tid + off];
        __syncthreads();
    }
    if (tid == 0) partials[b] = s_red[0];
}

// ---------------------------------------------------------------------------
// Kernel 1b: zero-pad the partials tail up to a RED_CHUNK multiple so the
// reduce loop needs no bounds checks (no EXEC-masked loads).
// ---------------------------------------------------------------------------
__global__ void pad_partials(float* __restrict__ partials, int n, int npad)
{
    int i = n + blockIdx.x * blockDim.x + threadIdx.x;
    if (i < npad) partials[i] = 0.0f;
}

// ---------------------------------------------------------------------------
// Kernel 2: single-wave WMMA reduction. A = ones(16x4); each lane contributes
// two ADJACENT floats to B (one global_load_b64, fully coalesced). With
// A = ones, D[m,n] += sum_k B[k,n] — any bijection of 64 values into B sums
// them all. Two accumulators keep two loads in flight per WMMA pair.
// Deterministic: fixed order, no atomics.
// ---------------------------------------------------------------------------
__global__ void __launch_bounds__(32)
wmma_reduce(const float* __restrict__ partials, float* __restrict__ out,
            int npad, float inv_batch)
{
    const int lane = threadIdx.x;
    v8f acc0 = {};
    v8f acc1 = {};
    v2f a;
    a[0] = 1.0f;
    a[1] = 1.0f;

    const v2f* p = (const v2f*)(partials) + lane;   // lane's 8-byte slot
    #pragma unroll 2
    for (int base = 0; base < npad; base += RED_CHUNK) {
        v2f b0 = p[0];       // elements [base      , base+ 64) across wave
        v2f b1 = p[32];      // elements [base +  64, base+128) across wave
        acc0 = __builtin_amdgcn_wmma_f32_16x16x4_f32(
            false, a, false, b0, (short)0, acc0, false, false);
        acc1 = __builtin_amdgcn_wmma_f32_16x16x4_f32(
            false, a, false, b1, (short)0, acc1, false, false);
        p += 64;             // RED_CHUNK floats
    }

    // acc[0]: lanes 0..15 = row M=0 (cols 0..15); lanes 16..31 duplicate (M=8)
    float v = acc0[0] + acc1[0];
    #pragma unroll
    for (int off = 8; off >= 1; off >>= 1)
        v += __shfl_xor(v, off, 16);

    if (lane == 0) out[0] = v * inv_batch;
}

// ---------------------------------------------------------------------------
extern "C" void kernel_launch(void* const* d_in, const int* in_sizes, int n_in,
                              void* d_out, int out_size, void* d_ws, size_t ws_size,
                              hipStream_t stream) {
    const float* outputs = (const float*)d_in[0];
    const float* target  = (const float*)d_in[1];
    float* out = (float*)d_out;
    float* ws  = (float*)d_ws;

    const int batch = in_sizes[0] / (GRID_N * FEAT);
    const int npad  = ((batch + RED_CHUNK - 1) / RED_CHUNK) * RED_CHUNK;

    yolo_loss_main<<<batch, 64, 0, stream>>>(outputs, target, ws, batch);
    if (npad > batch) {
        int tail = npad - batch;
        pad_partials<<<(tail + 63) / 64, 64, 0, stream>>>(ws, batch, npad);
    }
    wmma_reduce<<<1, 32, 0, stream>>>(ws, out, npad, 1.0f / (float)batch);
}